// MultitaskSNN_20529943675435
// MI455X (gfx1250) — compile-verified
//
#include <hip/hip_runtime.h>

// ---------------------------------------------------------------------------
// Multitask SNN forward, MI455X (gfx1250), fp32 WMMA path.
//   x:(T,B,32) -> [lin+LIF]x5 -> records (mco:3, spkc1:8, mro:1, spk1:28, spk2:14)
// One wave32 owns a 16-row batch tile for all T=100 steps (state in VGPRs).
// GEMMs via V_WMMA_F32_16X16X4_F32 (full fp32, matches reference numerics up
// to reassociation). Layer-to-layer operand transpose via per-wave LDS.
// ---------------------------------------------------------------------------

typedef __attribute__((ext_vector_type(2))) float v2f;
typedef __attribute__((ext_vector_type(8))) float v8f;

#define TT 100
#define BB 32768
#define DD 32
#define ROWSTRIDE 36   // LDS row stride (floats): bank-conflict-free, float4 aligned

__device__ __forceinline__ v8f wmma4(v2f a, v2f b, v8f c) {
  // D = A(16x4,f32) * B(4x16,f32) + C(16x16,f32)
  return __builtin_amdgcn_wmma_f32_16x16x4_f32(
      /*neg_a=*/false, a, /*neg_b=*/false, b,
      /*c_mod=*/(short)0, c, /*reuse_a=*/false, /*reuse_b=*/false);
}

// snnTorch Leaky, reset_mechanism='subtract', beta=0.9, thresh=1.0
__device__ __forceinline__ v8f lif(v8f cur, v8f& mem) {
  v8f spk;
#pragma unroll
  for (int i = 0; i < 8; ++i) {
    const float mprev = mem[i];
    const float reset = (mprev > 1.0f) ? 1.0f : 0.0f;
    const float nm    = 0.9f * mprev + cur[i] - reset;  // reset * THRESH
    mem[i] = nm;
    spk[i] = (nm > 1.0f) ? 1.0f : 0.0f;                 // (nm - 1 > 0)
  }
  return spk;
}

__device__ __forceinline__ v8f splat8(float s) {
  v8f r;
#pragma unroll
  for (int i = 0; i < 8; ++i) r[i] = s;
  return r;
}

__global__ __launch_bounds__(256) void snn_persistent_kernel(
    const float* __restrict__ x,
    const float* __restrict__ w_s1, const float* __restrict__ b_s1,
    const float* __restrict__ w_s2, const float* __restrict__ b_s2,
    const float* __restrict__ w_c1, const float* __restrict__ b_c1,
    const float* __restrict__ w_co, const float* __restrict__ b_co,
    const float* __restrict__ w_ro, const float* __restrict__ b_ro,
    float* __restrict__ out)
{
  __shared__ float stage[8][16 * ROWSTRIDE];   // per-wave private staging

  const int lane = threadIdx.x & 31;
  const int wave = threadIdx.x >> 5;
  const int half = lane >> 4;      // 0: lanes 0-15, 1: lanes 16-31
  const int lm   = lane & 15;      // N index (B/C/D layouts) / M row (A layout)
  const int rowbase = (blockIdx.x * 8 + wave) * 16;
  float* sbuf = &stage[wave][0];

  // ---- weight preload in B-operand layout -------------------------------
  // B(4x16) f32: VGPR v, lane L -> element [k = 4c + 2*(L/16) + v][n = L%16],
  // i.e. w[n][k], a contiguous float2 in the (out,in) row-major weight.
  auto wl = [&](const float* w, int n, int k, int N, int K) -> float {
    return (n < N && k < K) ? w[n * K + k] : 0.0f;
  };

  v2f Bs1[2][8];                        // 32 -> 28 (2 N-tiles, K=32: 8 chunks)
#pragma unroll
  for (int t2 = 0; t2 < 2; ++t2)
#pragma unroll
    for (int c = 0; c < 8; ++c) {
      const int n = lm + 16 * t2, k = 4 * c + 2 * half;
      Bs1[t2][c].x = wl(w_s1, n, k,     28, 32);
      Bs1[t2][c].y = wl(w_s1, n, k + 1, 28, 32);
    }
  v2f Bs2[7];                           // 28 -> 14 (K=28: 7 chunks)
#pragma unroll
  for (int c = 0; c < 7; ++c) {
    const int k = 4 * c + 2 * half;
    Bs2[c].x = wl(w_s2, lm, k,     14, 28);
    Bs2[c].y = wl(w_s2, lm, k + 1, 14, 28);
  }
  v2f Bc1[4], Bro[4];                   // 14 -> 8 and 14 -> 1 (K pad to 16)
#pragma unroll
  for (int c = 0; c < 4; ++c) {
    const int k = 4 * c + 2 * half;
    Bc1[c].x = wl(w_c1, lm, k,     8, 14);
    Bc1[c].y = wl(w_c1, lm, k + 1, 8, 14);
    Bro[c].x = wl(w_ro, lm, k,     1, 14);
    Bro[c].y = wl(w_ro, lm, k + 1, 1, 14);
  }
  v2f Bco[2];                           // 8 -> 3 (K=8: 2 chunks)
#pragma unroll
  for (int c = 0; c < 2; ++c) {
    const int k = 4 * c + 2 * half;
    Bco[c].x = wl(w_co, lm, k,     3, 8);
    Bco[c].y = wl(w_co, lm, k + 1, 3, 8);
  }

  const float bias_s1_0 = b_s1[lm];
  const float bias_s1_1 = (lm < 12) ? b_s1[16 + lm] : 0.0f;
  const float bias_s2   = (lm < 14) ? b_s2[lm] : 0.0f;
  const float bias_c1   = (lm <  8) ? b_c1[lm] : 0.0f;
  const float bias_co   = (lm <  3) ? b_co[lm] : 0.0f;
  const float bias_ro   = (lm == 0) ? b_ro[0] : 0.0f;

  // ---- output layout (tuple concatenated flat) --------------------------
  const size_t TB = (size_t)TT * BB;
  float* o_mco  = out;            // (T,B,3)  mem_class_rec
  float* o_spkc = out + TB * 3;   // (T,B,8)  spk_class1_rec
  float* o_mro  = out + TB * 11;  // (T,B,1)  mem_reg_rec
  float* o_spk1 = out + TB * 12;  // (T,B,28) spk1_rec
  float* o_spk2 = out + TB * 40;  // (T,B,14) spk2_rec

  // ---- membrane state (C/D layout: VGPR v, lane L -> m=v+8*half, n=lm) --
  v8f m1a = {}, m1b = {}, m2 = {}, mc1 = {}, mco = {}, mro = {};

  for (int t = 0; t < TT; ++t) {
    const float* xt = x + ((size_t)t * BB + rowbase) * DD;   // contiguous 2KB tile

    // stage x tile into LDS, fully coalesced (64B per lane)
    {
      const int r = lane >> 1, h = (lane & 1) * 16;
#pragma unroll
      for (int i = 0; i < 4; ++i) {
        const float4 v4 = *(const float4*)(xt + r * DD + h + 4 * i);
        *(float4*)(&sbuf[r * ROWSTRIDE + h + 4 * i]) = v4;
      }
    }
    if (t + 1 < TT)
      __builtin_prefetch(xt + (size_t)BB * DD + lane * 16, 0, 1);
    asm volatile("" ::: "memory");

    // ---- layer s1: 32 -> 28 --------------------------------------------
    v2f A1[8];
#pragma unroll
    for (int c = 0; c < 8; ++c)
      A1[c] = *(const v2f*)(&sbuf[lm * ROWSTRIDE + 4 * c + 2 * half]);
    v8f acc0 = splat8(bias_s1_0), acc1 = splat8(bias_s1_1);
#pragma unroll
    for (int c = 0; c < 8; ++c) acc0 = wmma4(A1[c], Bs1[0][c], acc0);
#pragma unroll
    for (int c = 0; c < 8; ++c) acc1 = wmma4(A1[c], Bs1[1][c], acc1);
    const v8f s1a = lif(acc0, m1a);
    const v8f s1b = lif(acc1, m1b);
#pragma unroll
    for (int v = 0; v < 8; ++v) {
      const int m = v + 8 * half;
      const size_t tb = (size_t)t * BB + rowbase + m;
      o_spk1[tb * 28 + lm] = s1a[v];
      sbuf[m * ROWSTRIDE + lm] = s1a[v];
      if (lm < 12) {
        o_spk1[tb * 28 + 16 + lm] = s1b[v];
        sbuf[m * ROWSTRIDE + 16 + lm] = s1b[v];
      }
    }
    asm volatile("" ::: "memory");

    // ---- layer s2: 28 -> 14 --------------------------------------------
    v2f A2[7];
#pragma unroll
    for (int c = 0; c < 7; ++c)
      A2[c] = *(const v2f*)(&sbuf[lm * ROWSTRIDE + 4 * c + 2 * half]);
    v8f acc2 = splat8(bias_s2);
#pragma unroll
    for (int c = 0; c < 7; ++c) acc2 = wmma4(A2[c], Bs2[c], acc2);
    const v8f s2 = lif(acc2, m2);   // lanes n>=14 are exact zeros (zero W, zero bias)
#pragma unroll
    for (int v = 0; v < 8; ++v) {
      const int m = v + 8 * half;
      const size_t tb = (size_t)t * BB + rowbase + m;
      if (lm < 14) o_spk2[tb * 14 + lm] = s2[v];
      sbuf[m * ROWSTRIDE + lm] = s2[v];
    }
    asm volatile("" ::: "memory");

    // ---- layers c1 (14->8) and ro (14->1) share the spk2 A operand ------
    v2f A3[4];
#pragma unroll
    for (int c = 0; c < 4; ++c)
      A3[c] = *(const v2f*)(&sbuf[lm * ROWSTRIDE + 4 * c + 2 * half]);
    v8f accC = splat8(bias_c1);
    v8f accR = splat8(bias_ro);
#pragma unroll
    for (int c = 0; c < 4; ++c) accC = wmma4(A3[c], Bc1[c], accC);
#pragma unroll
    for (int c = 0; c < 4; ++c) accR = wmma4(A3[c], Bro[c], accR);
    const v8f sc1 = lif(accC, mc1);
    (void)lif(accR, mro);           // spike discarded; mro recorded
#pragma unroll
    for (int v = 0; v < 8; ++v) {
      const int m = v + 8 * half;
      const size_t tb = (size_t)t * BB + rowbase + m;
      if (lm < 8) o_spkc[tb * 8 + lm] = sc1[v];
      if (lm == 0) o_mro[tb] = mro[v];
      sbuf[m * ROWSTRIDE + lm] = (lm < 8) ? sc1[v] : 0.0f;
    }
    asm volatile("" ::: "memory");

    // ---- layer co: 8 -> 3 ----------------------------------------------
    v2f A4[2];
#pragma unroll
    for (int c = 0; c < 2; ++c)
      A4[c] = *(const v2f*)(&sbuf[lm * ROWSTRIDE + 4 * c + 2 * half]);
    v8f accO = splat8(bias_co);
#pragma unroll
    for (int c = 0; c < 2; ++c) accO = wmma4(A4[c], Bco[c], accO);
    (void)lif(accO, mco);           // spike discarded; mco recorded
#pragma unroll
    for (int v = 0; v < 8; ++v) {
      const int m = v + 8 * half;
      const size_t tb = (size_t)t * BB + rowbase + m;
      if (lm < 3) o_mco[tb * 3 + lm] = mco[v];
    }
  }
}

extern "C" void kernel_launch(void* const* d_in, const int* in_sizes, int n_in,
                              void* d_out, int out_size, void* d_ws, size_t ws_size,
                              hipStream_t stream) {
  (void)in_sizes; (void)n_in; (void)out_size; (void)d_ws; (void)ws_size;
  const float* x    = (const float*)d_in[0];
  const float* w_s1 = (const float*)d_in[1];
  const float* b_s1 = (const float*)d_in[2];
  const float* w_s2 = (const float*)d_in[3];
  const float* b_s2 = (const float*)d_in[4];
  const float* w_c1 = (const float*)d_in[5];
  const float* b_c1 = (const float*)d_in[6];
  const float* w_co = (const float*)d_in[7];
  const float* b_co = (const float*)d_in[8];
  const float* w_ro = (const float*)d_in[9];
  const float* b_ro = (const float*)d_in[10];

  // 16 batch rows per wave, 8 waves per block -> 128 rows/block, 256 blocks.
  dim3 grid(BB / 128), block(256);
  snn_persistent_kernel<<<grid, block, 0, stream>>>(
      x, w_s1, b_s1, w_s2, b_s2, w_c1, b_c1, w_co, b_co, w_ro, b_ro,
      (float*)d_out);
}